// ConvCNPWeather_15358803051053
// MI455X (gfx1250) — compile-verified
//
#include <hip/hip_runtime.h>
#include <hip/hip_bf16.h>

#define BB   8
#define GXN  240
#define GYN  121
#define NCH  213

typedef __attribute__((ext_vector_type(2))) float v2f;
typedef __attribute__((ext_vector_type(8))) float v8f;

// internal grids: gx[i] = i/239, gy[j] = j/240 - 0.25
__device__ __forceinline__ float grid_x(int i){ return (float)i * (1.0f/239.0f); }
__device__ __forceinline__ float grid_y(int j){ return (float)j * (1.0f/240.0f) - 0.25f; }

__device__ __forceinline__ v8f wmma4(v2f a, v2f b, v8f c){
  // D = A(16x4 f32) * B(4x16 f32) + C(16x16 f32)
  return __builtin_amdgcn_wmma_f32_16x16x4_f32(false, a, false, b, (short)0, c, false, false);
}

// ---------------------------------------------------------------------------
// Off-grid setconv: per (b,c) two GEMMs den/sig = (kx*w)^T @ ky, K = N points.
// 3x2 register blocking: each wave owns a 48x32 output block (6 tile-pairs,
// 12 WMMAs per K=4 step against 10 v_exp_f32 + 3 b64 loads).
// grid: (5 i-macros, 4 j-macros, B*C), block = 1 wave
// ---------------------------------------------------------------------------
__global__ void __launch_bounds__(32)
offgrid_setconv(const float* __restrict__ px, const float* __restrict__ py,
                const float* __restrict__ yv, const float* __restrict__ ls,
                float* __restrict__ out,
                int N, int C,
                long pxStrideB, long pxStrideC,
                long yStrideB,  long yStrideC,
                int outChanBase)
{
  const int lane  = threadIdx.x & 31;
  const int tileI = blockIdx.x;          // 0..4 (48 rows each)
  const int tileJ = blockIdx.y;          // 0..3 (32 cols each)
  const int bc    = blockIdx.z;
  const int b = bc / C, c = bc % C;

  const float* pxp = px + (long)b*pxStrideB + (long)c*pxStrideC;
  const float* pyp = py + (long)b*pxStrideB + (long)c*pxStrideC;
  const float* yp  = yv + (long)b*yStrideB  + (long)c*yStrideC;

  const float lsv = ls[0];
  const float s   = -0.5f / (lsv*lsv);
  const int m16   = lane & 15;
  const int kHalf = (lane >> 4) << 1;    // lanes 0-15 -> K 0,1 ; lanes 16-31 -> K 2,3

  float gxi[3], gyj[2];
#pragma unroll
  for (int it = 0; it < 3; ++it) gxi[it] = grid_x(tileI*48 + it*16 + m16);
#pragma unroll
  for (int jt = 0; jt < 2; ++jt) gyj[jt] = grid_y(tileJ*32 + jt*16 + m16);

  v8f accD[3][2], accS[3][2];
#pragma unroll
  for (int it = 0; it < 3; ++it)
#pragma unroll
    for (int jt = 0; jt < 2; ++jt) { accD[it][jt] = (v8f){0,0,0,0,0,0,0,0};
                                     accS[it][jt] = (v8f){0,0,0,0,0,0,0,0}; }

  for (int n0 = 0; n0 < N; n0 += 4) {
    const int p0 = n0 + kHalf, p1 = p0 + 1;
    // speculative stream-ahead (safe past end: translation faults dropped)
    __builtin_prefetch(&pxp[p0 + 256], 0, 1);
    __builtin_prefetch(&pyp[p0 + 256], 0, 1);
    __builtin_prefetch(&yp [p0 + 256], 0, 1);

    const float x0 = pxp[p0], x1 = pxp[p1];
    const float q0 = pyp[p0], q1 = pyp[p1];
    const float y0 = yp[p0],  y1 = yp[p1];
    const int   f0 = __builtin_isfinite(y0), f1 = __builtin_isfinite(y1);
    const float m0 = f0 ? 1.0f : 0.0f,  m1 = f1 ? 1.0f : 0.0f;
    const float w0 = f0 ? y0  : 0.0f,   w1 = f1 ? y1  : 0.0f;

    v2f aD[3], aS[3], bk[2];
#pragma unroll
    for (int it = 0; it < 3; ++it) {
      const float d0 = x0 - gxi[it], d1 = x1 - gxi[it];
      const float kx0 = __expf(s*d0*d0), kx1 = __expf(s*d1*d1);
      aD[it] = (v2f){ kx0*m0, kx1*m1 };
      aS[it] = (v2f){ kx0*w0, kx1*w1 };
    }
#pragma unroll
    for (int jt = 0; jt < 2; ++jt) {
      const float e0 = q0 - gyj[jt], e1 = q1 - gyj[jt];
      bk[jt] = (v2f){ __expf(s*e0*e0), __expf(s*e1*e1) };
    }
#pragma unroll
    for (int it = 0; it < 3; ++it)
#pragma unroll
      for (int jt = 0; jt < 2; ++jt) {
        accD[it][jt] = wmma4(aD[it], bk[jt], accD[it][jt]);
        accS[it][jt] = wmma4(aS[it], bk[jt], accS[it][jt]);
      }
  }

  const int  mhi = (lane >> 4) * 8;
  const long chD = (long)b*NCH + outChanBase + 2*c;
#pragma unroll
  for (int jt = 0; jt < 2; ++jt) {
    const int jj = tileJ*32 + jt*16 + m16;
    if (jj < GYN) {
#pragma unroll
      for (int it = 0; it < 3; ++it)
        for (int r = 0; r < 8; ++r) {
          const int  ii   = tileI*48 + it*16 + mhi + r;
          const long idxD = (chD*GXN + ii)*(long)GYN + jj;
          const float dv  = accD[it][jt][r];
          out[idxD]                 = dv;
          out[idxD + (long)GXN*GYN] = accS[it][jt][r] / (dv + 1e-6f);
        }
    }
  }
}

// ---------------------------------------------------------------------------
// On-grid setconv stage 1:  t(i,w) = sum_h kx[h,i] * W[c,h,w]  (den:mask, sig:val)
// MODE 0: sat (B,C,H,Wd) plain finite mask
// MODE 1: amsua (B,Wd,H,C) with zeros->NaN and last channel NaN'd
// MODE 2: amsub/hirs (B,H,Wd,C) with zeros->NaN
// 3x2 blocking; grid: (5 i-macros, Wpad/32 w-macros, B*C)
// ---------------------------------------------------------------------------
template<int MODE>
__device__ __forceinline__ float loadW(const float* __restrict__ w,
                                       int b, int c, int h, int wc,
                                       int H, int Wd, int C)
{
  if (MODE == 0) return w[(((long)b*C  + c )*H  + h )*Wd + wc];
  if (MODE == 1) return w[(((long)b*Wd + wc)*H  + h )*C  + c ];
  /* MODE == 2 */return w[(((long)b*H  + h )*Wd + wc)*C  + c ];
}

template<int MODE>
__global__ void __launch_bounds__(32)
ongrid_stage1(const float* __restrict__ xlon, const float* __restrict__ w,
              const float* __restrict__ ls,
              float* __restrict__ tden, float* __restrict__ tsig,
              int H, int Wd, int Wpad, int C)
{
  const int lane  = threadIdx.x & 31;
  const int tileI = blockIdx.x;          // 0..4
  const int tileW = blockIdx.y;          // 0..Wpad/32-1
  const int bc    = blockIdx.z;
  const int b = bc / C, c = bc % C;

  const float* lon = xlon + (long)b*H;
  const float lsv = ls[0];
  const float s   = -0.5f / (lsv*lsv);
  const int m16   = lane & 15;
  const int kHalf = (lane >> 4) << 1;

  float gxi[3]; int wcol[2], wcc[2]; bool wok[2];
#pragma unroll
  for (int it = 0; it < 3; ++it) gxi[it] = grid_x(tileI*48 + it*16 + m16);
#pragma unroll
  for (int jt = 0; jt < 2; ++jt) {
    wcol[jt] = tileW*32 + jt*16 + m16;
    wok[jt]  = wcol[jt] < Wd;
    wcc[jt]  = wok[jt] ? wcol[jt] : (Wd - 1);   // clamped, branch-free load
  }

  v8f accD[3][2], accS[3][2];
#pragma unroll
  for (int it = 0; it < 3; ++it)
#pragma unroll
    for (int jt = 0; jt < 2; ++jt) { accD[it][jt] = (v8f){0,0,0,0,0,0,0,0};
                                     accS[it][jt] = (v8f){0,0,0,0,0,0,0,0}; }

  for (int h0 = 0; h0 < H; h0 += 4) {
    const int ha = h0 + kHalf, hb = ha + 1;
    const float la = lon[ha], lb = lon[hb];

    v2f A[3];
#pragma unroll
    for (int it = 0; it < 3; ++it) {
      const float da = la - gxi[it], db = lb - gxi[it];
      A[it] = (v2f){ __expf(s*da*da), __expf(s*db*db) };
    }

    v2f Bd[2], Bs[2];
#pragma unroll
    for (int jt = 0; jt < 2; ++jt) {
      float ra = loadW<MODE>(w, b, c, ha, wcc[jt], H, Wd, C);
      float rb = loadW<MODE>(w, b, c, hb, wcc[jt], H, Wd, C);
      bool fa = __builtin_isfinite(ra) && wok[jt];
      bool fb = __builtin_isfinite(rb) && wok[jt];
      if (MODE != 0) { fa = fa && (ra != 0.0f); fb = fb && (rb != 0.0f); }
      if (MODE == 1 && c == C-1) { fa = false; fb = false; }
      Bd[jt] = (v2f){ fa ? 1.0f : 0.0f, fb ? 1.0f : 0.0f };
      Bs[jt] = (v2f){ fa ? ra   : 0.0f, fb ? rb   : 0.0f };
    }
#pragma unroll
    for (int it = 0; it < 3; ++it)
#pragma unroll
      for (int jt = 0; jt < 2; ++jt) {
        accD[it][jt] = wmma4(A[it], Bd[jt], accD[it][jt]);
        accS[it][jt] = wmma4(A[it], Bs[jt], accS[it][jt]);
      }
  }

  const int mhi = (lane >> 4) * 8;
#pragma unroll
  for (int it = 0; it < 3; ++it)
#pragma unroll
    for (int jt = 0; jt < 2; ++jt)
      for (int r = 0; r < 8; ++r) {
        const int  ii  = tileI*48 + it*16 + mhi + r;
        const long idx = ((long)bc*GXN + ii)*(long)Wpad + wcol[jt]; // wcol < Wpad always
        tden[idx] = accD[it][jt][r];
        tsig[idx] = accS[it][jt][r];
      }
}

// ---------------------------------------------------------------------------
// On-grid setconv stage 2:  out(i,j) = sum_w t(i,w) * ky[w,j], normalize, store
// 3x2 blocking; grid: (5 i-macros, 4 j-macros, B*C)
// ---------------------------------------------------------------------------
__global__ void __launch_bounds__(32)
ongrid_stage2(const float* __restrict__ xlat,
              const float* __restrict__ tden, const float* __restrict__ tsig,
              const float* __restrict__ ls, float* __restrict__ out,
              int Wd, int Wpad, int C, int outChanBase)
{
  const int lane  = threadIdx.x & 31;
  const int tileI = blockIdx.x;          // 0..4
  const int tileJ = blockIdx.y;          // 0..3
  const int bc    = blockIdx.z;
  const int b = bc / C, c = bc % C;

  const float* lat = xlat + (long)b*Wd;
  const float lsv = ls[0];
  const float s   = -0.5f / (lsv*lsv);
  const int m16   = lane & 15;
  const int kHalf = (lane >> 4) << 1;

  float gyj[2]; long tbase[3];
#pragma unroll
  for (int jt = 0; jt < 2; ++jt) gyj[jt] = grid_y(tileJ*32 + jt*16 + m16);
#pragma unroll
  for (int it = 0; it < 3; ++it) {
    const int irow = tileI*48 + it*16 + m16;
    tbase[it] = (long)bc*GXN*Wpad + (long)irow*Wpad;
  }

  v8f accD[3][2], accS[3][2];
#pragma unroll
  for (int it = 0; it < 3; ++it)
#pragma unroll
    for (int jt = 0; jt < 2; ++jt) { accD[it][jt] = (v8f){0,0,0,0,0,0,0,0};
                                     accS[it][jt] = (v8f){0,0,0,0,0,0,0,0}; }

  for (int w0 = 0; w0 < Wpad; w0 += 4) {
    const int wa = w0 + kHalf, wb = wa + 1;
    const int wac = (wa < Wd) ? wa : (Wd - 1);   // clamped, branch-free
    const int wbc = (wb < Wd) ? wb : (Wd - 1);
    const float la = lat[wac], lb = lat[wbc];

    v2f Bk[2];
#pragma unroll
    for (int jt = 0; jt < 2; ++jt) {
      const float ea = la - gyj[jt], eb = lb - gyj[jt];
      const float kya = (wa < Wd) ? __expf(s*ea*ea) : 0.0f;
      const float kyb = (wb < Wd) ? __expf(s*eb*eb) : 0.0f;
      Bk[jt] = (v2f){ kya, kyb };
    }

    v2f Ad[3], As[3];
#pragma unroll
    for (int it = 0; it < 3; ++it) {
      Ad[it] = (v2f){ tden[tbase[it] + wa], tden[tbase[it] + wb] };
      As[it] = (v2f){ tsig[tbase[it] + wa], tsig[tbase[it] + wb] };
    }
#pragma unroll
    for (int it = 0; it < 3; ++it)
#pragma unroll
      for (int jt = 0; jt < 2; ++jt) {
        accD[it][jt] = wmma4(Ad[it], Bk[jt], accD[it][jt]);
        accS[it][jt] = wmma4(As[it], Bk[jt], accS[it][jt]);
      }
  }

  const int  mhi = (lane >> 4) * 8;
  const long chD = (long)b*NCH + outChanBase + 2*c;
#pragma unroll
  for (int jt = 0; jt < 2; ++jt) {
    const int jj = tileJ*32 + jt*16 + m16;
    if (jj < GYN) {
#pragma unroll
      for (int it = 0; it < 3; ++it)
        for (int r = 0; r < 8; ++r) {
          const int  ii   = tileI*48 + it*16 + mhi + r;
          const long idxD = (chD*GXN + ii)*(long)GYN + jj;
          const float dv  = accD[it][jt][r];
          out[idxD]                 = dv;
          out[idxD + (long)GXN*GYN] = accS[it][jt][r] / (dv + 1e-6f);
        }
    }
  }
}

// ---------------------------------------------------------------------------
// Elementwise channels
// ---------------------------------------------------------------------------
__global__ void resize_flip_nan0(const float* __restrict__ in, float* __restrict__ out,
                                 int C, int outChanBase)
{
  // in: (B,120,60,C); out[b, base+c, i, j] = nan0(in[b, i/2, ((120-j)*60)/121, c])
  long idx = (long)blockIdx.x*blockDim.x + threadIdx.x;
  const long total = (long)BB*C*GXN*GYN;
  if (idx >= total) return;
  int j = idx % GYN; long t = idx / GYN;
  int i = t % GXN; t /= GXN;
  int c = t % C;  int b = t / C;
  const int hi = i >> 1;
  const int wi = ((120 - j) * 60) / 121;
  float v = in[(((long)b*120 + hi)*60 + wi)*C + c];
  if (!__builtin_isfinite(v)) v = 0.0f;
  out[(((long)b*NCH + outChanBase + c)*GXN + i)*(long)GYN + j] = v;
}

__global__ void elev_kernel(const float* __restrict__ in, float* __restrict__ out)
{
  // out[b,183,i,j] = era5_elev[b,0,2i,241-2j]
  long idx = (long)blockIdx.x*blockDim.x + threadIdx.x;
  const long total = (long)BB*GXN*GYN;
  if (idx >= total) return;
  int j = idx % GYN; long t = idx / GYN;
  int i = t % GXN;  int b = t / GXN;
  float v = in[((long)b*480 + 2*i)*242 + (241 - 2*j)];
  out[(((long)b*NCH + 183)*GXN + i)*(long)GYN + j] = v;
}

__global__ void clim_kernel(const float* __restrict__ in, float* __restrict__ out)
{
  long idx = (long)blockIdx.x*blockDim.x + threadIdx.x;
  const long total = (long)BB*24*GXN*GYN;
  if (idx >= total) return;
  long sp = idx % ((long)GXN*GYN); long t = idx / ((long)GXN*GYN);
  int c = t % 24; int b = t / 24;
  out[(((long)b*NCH + 184 + c)*GXN)*(long)GYN + sp] = in[(((long)b*24 + c)*GXN)*(long)GYN + sp];
}

__global__ void aux_kernel(const float* __restrict__ aux, float* __restrict__ out)
{
  long idx = (long)blockIdx.x*blockDim.x + threadIdx.x;
  const long total = (long)BB*5*GXN*GYN;
  if (idx >= total) return;
  long sp = idx % ((long)GXN*GYN); long t = idx / ((long)GXN*GYN);
  int k = t % 5; int b = t / 5;
  out[(((long)b*NCH + 208 + k)*GXN)*(long)GYN + sp] = aux[b*5 + k];
}

// ---------------------------------------------------------------------------
extern "C" void kernel_launch(void* const* d_in, const int* in_sizes, int n_in,
                              void* d_out, int out_size, void* d_ws, size_t ws_size,
                              hipStream_t stream)
{
  const float* iasi      = (const float*)d_in[0];
  const float* ascat     = (const float*)d_in[1];
  const float* hadisd_x  = (const float*)d_in[2];   // (4,B,2,5000)
  const float* hadisd_y  = (const float*)d_in[3];   // (4,B,5000)
  const float* icoads_x  = (const float*)d_in[4];   // (B,2,20000)
  const float* icoads_c  = (const float*)d_in[5];   // (B,5,20000)
  const float* sat_lon   = (const float*)d_in[6];   // (B,360)
  const float* sat_lat   = (const float*)d_in[7];   // (B,180)
  const float* sat_cur   = (const float*)d_in[8];   // (B,2,360,180)
  const float* amsua_lon = (const float*)d_in[9];   // (B,240)
  const float* amsua_lat = (const float*)d_in[10];  // (B,121)
  const float* amsua_cur = (const float*)d_in[11];  // (B,121,240,13)
  const float* amsub_lon = (const float*)d_in[12];
  const float* amsub_lat = (const float*)d_in[13];
  const float* amsub_cur = (const float*)d_in[14];  // (B,240,121,12)
  const float* hirs_lon  = (const float*)d_in[15];
  const float* hirs_lat  = (const float*)d_in[16];
  const float* hirs_cur  = (const float*)d_in[17];  // (B,240,121,26)
  const float* igra_x    = (const float*)d_in[18];  // (B,2,3000)
  const float* igra_c    = (const float*)d_in[19];  // (B,24,3000)
  const float* elev      = (const float*)d_in[20];  // (B,1,480,242)
  const float* clim      = (const float*)d_in[21];  // (B,24,240,121)
  const float* aux       = (const float*)d_in[22];  // (B,5)
  const float* ls        = (const float*)d_in[23];  // (1,)
  float* out = (float*)d_out;

  // workspace: stage-1 intermediates, reused sequentially (stream order).
  // max: HIRS = B*26*240*128 floats per buffer -> 51.1 MB total.
  const size_t maxT = (size_t)BB*26*GXN*128;
  float* tden = (float*)d_ws;
  float* tsig = tden + maxT;

  // --- elementwise channels ---
  { long n = (long)BB*8*GXN*GYN;
    resize_flip_nan0<<<dim3((n+255)/256),256,0,stream>>>(iasi, out, 8, 0); }
  { long n = (long)BB*3*GXN*GYN;
    resize_flip_nan0<<<dim3((n+255)/256),256,0,stream>>>(ascat, out, 3, 8); }
  { long n = (long)BB*GXN*GYN;
    elev_kernel<<<dim3((n+255)/256),256,0,stream>>>(elev, out); }
  { long n = (long)BB*24*GXN*GYN;
    clim_kernel<<<dim3((n+255)/256),256,0,stream>>>(clim, out); }
  { long n = (long)BB*5*GXN*GYN;
    aux_kernel<<<dim3((n+255)/256),256,0,stream>>>(aux, out); }

  // --- off-grid setconvs (WMMA f32 16x16x4, 3x2 blocked) ---
  // HadISD: C=4, N=5000; px stride over c = B*2*5000, over b = 2*5000
  offgrid_setconv<<<dim3(5,4,BB*4),32,0,stream>>>(
      hadisd_x, hadisd_x + 5000, hadisd_y, ls, out,
      5000, 4, 2L*5000, (long)BB*2*5000, 5000L, (long)BB*5000, 11);
  // ICOADS: C=5, N=20000
  offgrid_setconv<<<dim3(5,4,BB*5),32,0,stream>>>(
      icoads_x, icoads_x + 20000, icoads_c, ls, out,
      20000, 5, 2L*20000, 0L, 5L*20000, 20000L, 19);
  // IGRA: C=24, N=3000
  offgrid_setconv<<<dim3(5,4,BB*24),32,0,stream>>>(
      igra_x, igra_x + 3000, igra_c, ls, out,
      3000, 24, 2L*3000, 0L, 24L*3000, 3000L, 83);

  // --- on-grid setconvs, separable 2-stage ---
  // GRIDSAT: C=2, H=360, Wd=180, Wpad=192
  ongrid_stage1<0><<<dim3(5,6,BB*2),32,0,stream>>>(sat_lon, sat_cur, ls, tden, tsig, 360, 180, 192, 2);
  ongrid_stage2   <<<dim3(5,4,BB*2),32,0,stream>>>(sat_lat, tden, tsig, ls, out, 180, 192, 2, 29);
  // AMSU-A: C=13, H=240, Wd=121, Wpad=128, zeros+last-channel NaN'd
  ongrid_stage1<1><<<dim3(5,4,BB*13),32,0,stream>>>(amsua_lon, amsua_cur, ls, tden, tsig, 240, 121, 128, 13);
  ongrid_stage2   <<<dim3(5,4,BB*13),32,0,stream>>>(amsua_lat, tden, tsig, ls, out, 121, 128, 13, 33);
  // AMSU-B: C=12
  ongrid_stage1<2><<<dim3(5,4,BB*12),32,0,stream>>>(amsub_lon, amsub_cur, ls, tden, tsig, 240, 121, 128, 12);
  ongrid_stage2   <<<dim3(5,4,BB*12),32,0,stream>>>(amsub_lat, tden, tsig, ls, out, 121, 128, 12, 59);
  // HIRS: C=26
  ongrid_stage1<2><<<dim3(5,4,BB*26),32,0,stream>>>(hirs_lon, hirs_cur, ls, tden, tsig, 240, 121, 128, 26);
  ongrid_stage2   <<<dim3(5,4,BB*26),32,0,stream>>>(hirs_lat, tden, tsig, ls, out, 121, 128, 26, 131);
}